// PointVoxelizer_84155589198507
// MI455X (gfx1250) — compile-verified
//
#include <hip/hip_runtime.h>
#include <stdint.h>

// ---- problem constants (match reference: grid = (1000,1000,80), 4 batches) ----
#define GXD   1000
#define GYD   1000
#define GZD   80
#define YZD   (GYD * GZD)          // 80,000
#define GXYZD (GXD * YZD)          // 80,000,000 cells per batch
#define NBATCH 4
#define TOTAL_CELLS (4u * 80000000u)      // 320,000,000
#define WORDS       (TOTAL_CELLS / 32u)   // 10,000,000 u32 words (40 MB, fits L2)
#define WPB   8192u                        // words per scan block
#define NBLK  ((WORDS + WPB - 1u) / WPB)   // 1221

// ============================ pass 0: init ============================
__global__ void k_clear_mask(uint32_t* __restrict__ mask) {
  uint32_t i = blockIdx.x * blockDim.x + threadIdx.x;
  uint32_t stride = gridDim.x * blockDim.x;
  for (; i < WORDS; i += stride) mask[i] = 0u;
}

// coords region [0,4n) := -1 (0xFFFFFFFF), features region [4n,7n) := 0.0f
__global__ void k_init_out(uint32_t* __restrict__ out, int n) {
  uint32_t nco = 4u * (uint32_t)n;
  uint32_t tot = 7u * (uint32_t)n;
  uint32_t i = blockIdx.x * blockDim.x + threadIdx.x;
  uint32_t stride = gridDim.x * blockDim.x;
  for (; i < tot; i += stride) out[i] = (i < nco) ? 0xFFFFFFFFu : 0u;
}

// ============================ pass 1: bin points ============================
// Stages each block's 256 points (3072 B) into LDS with CDNA5 async
// global->LDS copies (ASYNCcnt path), then computes voxel hash + atomicOr bit.
__global__ __launch_bounds__(256) void k_bin(const float* __restrict__ pts,
                                             const int* __restrict__ bidx,
                                             uint32_t* __restrict__ mask,
                                             uint32_t* __restrict__ hashes,
                                             int n) {
  __shared__ __align__(16) float tile[256 * 3];
  int t = threadIdx.x;
  long long blockStart = (long long)blockIdx.x * 256;

  // 192 lanes x 16B = 3072B tile; all addresses 16B-aligned (256*12 = 3072).
  if (t < 192) {
    uint32_t ldsOff = (uint32_t)(uintptr_t)(&tile[0]) + (uint32_t)t * 16u;
    long long byteOff = blockStart * 12ll + (long long)t * 16ll;
    long long maxOff = (long long)n * 12ll - 16ll;   // stays 16B-aligned
    if (byteOff > maxOff) byteOff = maxOff;
    uint32_t gOff = (uint32_t)byteOff;
    asm volatile("global_load_async_to_lds_b128 %0, %1, %2"
                 :: "v"(ldsOff), "v"(gOff), "s"(pts)
                 : "memory");
  }
  asm volatile("s_wait_asynccnt 0" ::: "memory");
  __syncthreads();

  int i = (int)blockStart + t;
  if (i >= n) return;

  float px = tile[t * 3 + 0];
  float py = tile[t * 3 + 1];
  float pz = tile[t * 3 + 2];

  // exactly mirror reference f32 math: floor((p - pmin) / 0.1f)
  float qx = __fdiv_rn(__fsub_rn(px, -50.0f), 0.1f);
  float qy = __fdiv_rn(__fsub_rn(py, -50.0f), 0.1f);
  float qz = __fdiv_rn(__fsub_rn(pz,  -5.0f), 0.1f);
  int vx = (int)floorf(qx);
  int vy = (int)floorf(qy);
  int vz = (int)floorf(qz);

  bool valid = (vx >= 0) & (vx < GXD) & (vy >= 0) & (vy < GYD) &
               (vz >= 0) & (vz < GZD);

  uint32_t h = 0xFFFFFFFFu;
  if (valid) {
    int b = bidx[i];
    h = (uint32_t)b * (uint32_t)GXYZD +
        (uint32_t)vx * (uint32_t)YZD +
        (uint32_t)vy * (uint32_t)GZD + (uint32_t)vz;
    atomicOr((unsigned int*)&mask[h >> 5], 1u << (h & 31u));
  }
  hashes[i] = h;
}

// ===================== pass 2: per-chunk popcount sums =====================
__global__ __launch_bounds__(256) void k_block_sums(const uint32_t* __restrict__ mask,
                                                    uint32_t* __restrict__ blockSums) {
  __shared__ uint32_t s[256];
  uint32_t b = blockIdx.x, t = threadIdx.x;
  uint32_t start = b * WPB + t * 32u;
  uint32_t sum = 0u;
#pragma unroll
  for (int k = 0; k < 32; ++k) {
    uint32_t w = start + (uint32_t)k;
    sum += (w < WORDS) ? (uint32_t)__popc(mask[w]) : 0u;
  }
  s[t] = sum;
  __syncthreads();
  for (int off = 128; off > 0; off >>= 1) {
    if ((int)t < off) s[t] += s[t + off];
    __syncthreads();
  }
  if (t == 0) blockSums[b] = s[0];
}

// ========== pass 3: exclusive scan of block sums (Blelloch, 2048) ==========
__global__ __launch_bounds__(1024) void k_scan(uint32_t* __restrict__ blockSums,
                                               int* __restrict__ out_n) {
  __shared__ uint32_t s[2048];
  int t = threadIdx.x;
  s[t]        = ((uint32_t)t        < NBLK) ? blockSums[t]        : 0u;
  s[t + 1024] = ((uint32_t)t + 1024 < NBLK) ? blockSums[t + 1024] : 0u;
  int offset = 1;
  for (int d = 1024; d > 0; d >>= 1) {
    __syncthreads();
    if (t < d) {
      int ai = offset * (2 * t + 1) - 1;
      int bi = offset * (2 * t + 2) - 1;
      s[bi] += s[ai];
    }
    offset <<= 1;
  }
  __syncthreads();
  if (t == 0) {
    *out_n = (int)s[2047];   // n_voxels (int32 bit pattern into d_out)
    s[2047] = 0u;
  }
  for (int d = 1; d < 2048; d <<= 1) {
    offset >>= 1;
    __syncthreads();
    if (t < d) {
      int ai = offset * (2 * t + 1) - 1;
      int bi = offset * (2 * t + 2) - 1;
      uint32_t tmp = s[ai];
      s[ai] = s[bi];
      s[bi] += tmp;
    }
  }
  __syncthreads();
  if ((uint32_t)t < NBLK)        blockSums[t]        = s[t];
  if ((uint32_t)t + 1024 < NBLK) blockSums[t + 1024] = s[t + 1024];
}

// ================= pass 4: per-word exclusive popcount prefix =================
__global__ __launch_bounds__(256) void k_word_prefix(const uint32_t* __restrict__ mask,
                                                     const uint32_t* __restrict__ blockOff,
                                                     uint32_t* __restrict__ wordPrefix) {
  __shared__ uint32_t s[256];
  uint32_t b = blockIdx.x, t = threadIdx.x;
  uint32_t start = b * WPB + t * 32u;
  uint32_t sum = 0u;
#pragma unroll
  for (int k = 0; k < 32; ++k) {
    uint32_t w = start + (uint32_t)k;
    sum += (w < WORDS) ? (uint32_t)__popc(mask[w]) : 0u;
  }
  s[t] = sum;
  __syncthreads();
  // Hillis-Steele inclusive scan over 256 thread sums
  for (int off = 1; off < 256; off <<= 1) {
    uint32_t v = ((int)t >= off) ? s[t - off] : 0u;
    __syncthreads();
    s[t] += v;
    __syncthreads();
  }
  uint32_t run = blockOff[b] + s[t] - sum;   // exclusive prefix for this thread
#pragma unroll
  for (int k = 0; k < 32; ++k) {
    uint32_t w = start + (uint32_t)k;
    if (w < WORDS) {
      wordPrefix[w] = run;
      run += (uint32_t)__popc(mask[w]);
    }
  }
}

// ================== pass 5: emit unique voxels in sorted order ==================
__global__ void k_emit(const uint32_t* __restrict__ mask,
                       const uint32_t* __restrict__ wordPrefix,
                       int* __restrict__ out_coords,
                       float* __restrict__ out_feat,
                       int n) {
  uint32_t i = blockIdx.x * blockDim.x + threadIdx.x;
  uint32_t stride = gridDim.x * blockDim.x;
  for (uint32_t w = i; w < WORDS; w += stride) {
    uint32_t m = mask[w];
    if (!m) continue;
    uint32_t r = wordPrefix[w];
    do {
      uint32_t bit = (uint32_t)__builtin_ctz(m);
      m &= m - 1u;
      uint32_t c = (w << 5) + bit;                 // cell hash, < 3.2e8
      uint32_t bb  = c / (uint32_t)GXYZD;
      uint32_t rem = c % (uint32_t)GXYZD;
      uint32_t x  = rem / (uint32_t)YZD;
      uint32_t r2 = rem % (uint32_t)YZD;
      uint32_t y  = r2 / (uint32_t)GZD;
      uint32_t z  = r2 % (uint32_t)GZD;
      if (r < (uint32_t)n) {
        int* cr = out_coords + 4u * r;
        cr[0] = (int)bb; cr[1] = (int)z; cr[2] = (int)y; cr[3] = (int)x;
        float* fr = out_feat + 3u * r;
        fr[0] = __fadd_rn(-50.0f, __fmul_rn(__fadd_rn((float)x, 0.5f), 0.1f));
        fr[1] = __fadd_rn(-50.0f, __fmul_rn(__fadd_rn((float)y, 0.5f), 0.1f));
        fr[2] = __fadd_rn( -5.0f, __fmul_rn(__fadd_rn((float)z, 0.5f), 0.1f));
      }
      ++r;
    } while (m);
  }
}

// ==================== pass 6: point -> voxel rank lookup ====================
__global__ void k_map(const uint32_t* __restrict__ hashes,
                      const uint32_t* __restrict__ mask,
                      const uint32_t* __restrict__ wordPrefix,
                      int* __restrict__ out_map,
                      int n) {
  int i = blockIdx.x * blockDim.x + threadIdx.x;
  if (i >= n) return;
  int pf = i + 8192; if (pf >= n) pf = n - 1;
  __builtin_prefetch(&hashes[pf], 0, 0);          // global_prefetch_b8
  uint32_t h = hashes[i];
  if (h == 0xFFFFFFFFu) { out_map[i] = -1; return; }
  uint32_t w = h >> 5, b = h & 31u;
  uint32_t low = (b == 0u) ? 0u : (mask[w] & ((1u << b) - 1u));
  out_map[i] = (int)(wordPrefix[w] + (uint32_t)__popc(low));
}

// ================================ launcher ================================
extern "C" void kernel_launch(void* const* d_in, const int* in_sizes, int n_in,
                              void* d_out, int out_size, void* d_ws, size_t ws_size,
                              hipStream_t stream) {
  const float* pts = (const float*)d_in[0];
  const int*   bidx = (const int*)d_in[1];
  const int n = in_sizes[0] / 3;                  // 2,000,000

  // workspace layout (all regions fully rewritten every call -> deterministic)
  uint8_t* ws = (uint8_t*)d_ws;
  uint32_t* mask       = (uint32_t*)(ws);                                   // 40 MB
  uint32_t* wordPrefix = (uint32_t*)(ws + (size_t)WORDS * 4u);              // 40 MB
  uint32_t* hashes     = (uint32_t*)(ws + (size_t)WORDS * 8u);              // 4n B
  uint32_t* blockSums  = (uint32_t*)(ws + (size_t)WORDS * 8u + (size_t)n * 4u);

  // d_out: [coords 4n int][features 3n f32][map n int][n_voxels 1 int]
  uint32_t* out_u      = (uint32_t*)d_out;
  int*      out_coords = (int*)d_out;
  float*    out_feat   = (float*)d_out + (size_t)4 * n;
  int*      out_map    = (int*)d_out + (size_t)7 * n;
  int*      out_n      = (int*)d_out + (size_t)8 * n;

  const int TPB = 256;
  k_clear_mask <<<2048, TPB, 0, stream>>>(mask);
  k_init_out   <<<2048, TPB, 0, stream>>>(out_u, n);
  k_bin        <<<(n + TPB - 1) / TPB, TPB, 0, stream>>>(pts, bidx, mask, hashes, n);
  k_block_sums <<<NBLK, TPB, 0, stream>>>(mask, blockSums);
  k_scan       <<<1, 1024, 0, stream>>>(blockSums, out_n);
  k_word_prefix<<<NBLK, TPB, 0, stream>>>(mask, blockSums, wordPrefix);
  k_emit       <<<2048, TPB, 0, stream>>>(mask, wordPrefix, out_coords, out_feat, n);
  k_map        <<<(n + TPB - 1) / TPB, TPB, 0, stream>>>(hashes, mask, wordPrefix, out_map, n);
}